// GNN_45509473468603
// MI455X (gfx1250) — compile-verified
//
#include <hip/hip_runtime.h>

// GNN: 2x GCNConv(relu) + global mean pool + FC, fp32 end-to-end.
// GEMMs: V_WMMA_F32_16X16X4_F32. Per block: 32 M-rows staged in LDS, (N/64) waves.
// Each wave computes a 32x64 panel = 2 M-subtiles x 4 N-tiles (8 accumulators):
// per k-step -> 2 ds fragment reads + 8 B loads + 8 WMMAs (1 VMEM load per WMMA).
// Scatter/gather over edges uses float4 loads + global f32 atomics (L2-resident).

#define F_IN  128
#define HID   256
#define OUT_F 128
#define KMAX  256
#define LPAD  4                 // LDS row pad (floats): bank step 4 -> conflict-free frag reads

typedef __attribute__((ext_vector_type(2))) float v2f;
typedef __attribute__((ext_vector_type(8))) float v8f;

// ---------------- elementwise / scatter kernels ----------------

__global__ void zero_kernel(float* __restrict__ p, int n) {
  int t = blockIdx.x * blockDim.x + threadIdx.x;
  if (t < n) p[t] = 0.0f;
}

__global__ void deg_edges_kernel(const int* __restrict__ dst, float* __restrict__ deg, int E) {
  int e = blockIdx.x * blockDim.x + threadIdx.x;
  if (e < E) atomicAdd(&deg[dst[e]], 1.0f);
}

__global__ void dinv_kernel(float* __restrict__ deg, int n) {
  int i = blockIdx.x * blockDim.x + threadIdx.x;
  if (i < n) deg[i] = rsqrtf(deg[i] + 1.0f);   // +1 for self loop; deg >= 1 always
}

// agg[i,:] = h[i,:] * dinv[i]^2   (self-loop contribution; full overwrite)
__global__ void self_init_kernel(const float* __restrict__ h, const float* __restrict__ dinv,
                                 float* __restrict__ agg, int n) {
  int t  = blockIdx.x * blockDim.x + threadIdx.x;
  int i  = t >> 6;               // 64 float4-chunks per node (HID=256)
  int f4 = (t & 63) << 2;
  if (i >= n) return;
  float di = dinv[i];
  float w  = di * di;
  const float4 hv = *reinterpret_cast<const float4*>(h + (size_t)i * HID + f4);
  float4 o; o.x = hv.x * w; o.y = hv.y * w; o.z = hv.z * w; o.w = hv.w * w;
  *reinterpret_cast<float4*>(agg + (size_t)i * HID + f4) = o;
}

// agg[dst,:] += h[src,:] * dinv[src]*dinv[dst]   (per-edge scatter-add)
__global__ void edge_scatter_kernel(const float* __restrict__ h, const float* __restrict__ dinv,
                                    const int* __restrict__ src, const int* __restrict__ dst,
                                    float* __restrict__ agg, int E) {
  int t  = blockIdx.x * blockDim.x + threadIdx.x;
  int e  = t >> 6;
  int f4 = (t & 63) << 2;
  if (e >= E) return;
  int s = src[e], d = dst[e];
  float nrm = dinv[s] * dinv[d];
  const float4 hv = *reinterpret_cast<const float4*>(h + (size_t)s * HID + f4);
  float* ap = agg + (size_t)d * HID + f4;
  atomicAdd(ap + 0, hv.x * nrm);
  atomicAdd(ap + 1, hv.y * nrm);
  atomicAdd(ap + 2, hv.z * nrm);
  atomicAdd(ap + 3, hv.w * nrm);
}

__global__ void bias_relu_kernel(float* __restrict__ a, const float* __restrict__ bias, int total) {
  int t = blockIdx.x * blockDim.x + threadIdx.x;
  if (t >= total) return;
  float v = a[t] + bias[t & (HID - 1)];
  a[t] = v > 0.0f ? v : 0.0f;
}

// pooled[batch[i],:] += h[i,:];  cnt[batch[i]] += 1
__global__ void pool_scatter_kernel(const float* __restrict__ h, const int* __restrict__ batch,
                                    float* __restrict__ pooled, float* __restrict__ cnt, int n) {
  int t  = blockIdx.x * blockDim.x + threadIdx.x;
  int i  = t >> 6;
  int f4 = (t & 63) << 2;
  if (i >= n) return;
  int g = batch[i];
  if (f4 == 0) atomicAdd(&cnt[g], 1.0f);
  const float4 hv = *reinterpret_cast<const float4*>(h + (size_t)i * HID + f4);
  float* pp = pooled + (size_t)g * HID + f4;
  atomicAdd(pp + 0, hv.x);
  atomicAdd(pp + 1, hv.y);
  atomicAdd(pp + 2, hv.z);
  atomicAdd(pp + 3, hv.w);
}

__global__ void pool_div_kernel(float* __restrict__ pooled, const float* __restrict__ cnt, int total) {
  int t = blockIdx.x * blockDim.x + threadIdx.x;
  if (t >= total) return;
  float c = cnt[t >> 8];                 // HID = 256 features per graph
  pooled[t] = pooled[t] / fmaxf(c, 1.0f);
}

// ---------------- fp32 WMMA GEMM: C[M,N] = A[M,K] @ B[K,N] (+bias) ----------------
// Block covers 32 M-rows and all N columns: (N/64) waves. The 32xK A tile is staged
// once in LDS (float4 cooperative load); fragments read back as float2 (ds_load
// 2addr), row pad of 4 floats -> 16 distinct banks for the 16 lane-rows.
// Each wave: 2 M-subtiles x 4 N-tiles; each B fragment feeds 2 WMMAs, each A
// fragment feeds 4 WMMAs.
//
// A-frag (16x4 f32, 2 VGPRs): lane L<16 -> row=L, K={k0,k0+1}; lanes>=16 -> K={k0+2,k0+3}.
// B-frag (4x16 f32, 2 VGPRs): mirrored: lane half selects K pair, lane%16 = column.
// C/D  (16x16 f32, 8 VGPRs): VGPR r, lanes0-15 -> M=r, lanes16-31 -> M=r+8; N=lane%16.
__global__ void wmma_gemm_kernel(const float* __restrict__ A, const float* __restrict__ B,
                                 float* __restrict__ C, const float* __restrict__ bias,
                                 int M, int N, int K) {
  __shared__ float aLds[32 * (KMAX + LPAD)];
  const int ldaL = K + LPAD;

  // --- cooperative stage of the 32xK A tile into LDS (float4 granularity) ---
  const int m0 = blockIdx.x * 32;
  const int nThreads = blockDim.x;
  for (int idx = threadIdx.x * 4; idx < 32 * K; idx += nThreads * 4) {
    int r = idx / K;
    int c = idx - r * K;                       // multiple of 4 (K is a multiple of 4)
    const float4 av = *reinterpret_cast<const float4*>(A + (size_t)(m0 + r) * K + c);
    *reinterpret_cast<float4*>(&aLds[r * ldaL + c]) = av;
  }
  __syncthreads();

  const int wave = threadIdx.x >> 5;
  const int lane = threadIdx.x & 31;
  const int n0   = wave * 64;                  // 64-column panel per wave
  const int row  = lane & 15;
  const int hi   = (lane >> 4) << 1;           // 0 or 2: K-pair selector for lane half

  v8f acc[2][4];
#pragma unroll
  for (int m = 0; m < 2; ++m)
#pragma unroll
    for (int j = 0; j < 4; ++j) acc[m][j] = (v8f){};

  for (int k0 = 0; k0 < K; k0 += 4) {
    // two A fragments (M-subtiles 0 and 1) from LDS
    const float2 af0 = *reinterpret_cast<const float2*>(&aLds[row * ldaL + k0 + hi]);
    const float2 af1 = *reinterpret_cast<const float2*>(&aLds[(row + 16) * ldaL + k0 + hi]);
    v2f a0; a0.x = af0.x; a0.y = af0.y;
    v2f a1; a1.x = af1.x; a1.y = af1.y;

    const float* bp = B + (size_t)(k0 + hi) * N + n0 + row;
    v2f b[4];
#pragma unroll
    for (int j = 0; j < 4; ++j) { b[j].x = bp[j * 16]; b[j].y = bp[N + j * 16]; }

#pragma unroll
    for (int j = 0; j < 4; ++j) {
      acc[0][j] = __builtin_amdgcn_wmma_f32_16x16x4_f32(false, a0, false, b[j], (short)0, acc[0][j], false, false);
      acc[1][j] = __builtin_amdgcn_wmma_f32_16x16x4_f32(false, a1, false, b[j], (short)0, acc[1][j], false, false);
    }
  }

  // --- store 8 tiles (+optional bias) ---
  const int crow = (lane >> 4) * 8;            // M offset of VGPR0 for this lane half
  const int ccol = lane & 15;
#pragma unroll
  for (int m = 0; m < 2; ++m) {
    float* cbase = C + (size_t)(m0 + m * 16 + crow) * N + n0 + ccol;
#pragma unroll
    for (int j = 0; j < 4; ++j) {
      const float bv = bias ? bias[n0 + j * 16 + ccol] : 0.0f;
      float* cp = cbase + j * 16;
#pragma unroll
      for (int r = 0; r < 8; ++r) cp[(size_t)r * N] = acc[m][j][r] + bv;
    }
  }
}

// ---------------- host-side orchestration ----------------

static inline int ceil_div(long long a, long long b) { return (int)((a + b - 1) / b); }

extern "C" void kernel_launch(void* const* d_in, const int* in_sizes, int n_in,
                              void* d_out, int out_size, void* d_ws, size_t ws_size,
                              hipStream_t stream) {
  const float* x    = (const float*)d_in[0];
  const int*   eidx = (const int*)  d_in[1];
  const int*   batch= (const int*)  d_in[2];
  const float* W1   = (const float*)d_in[3];
  const float* b1   = (const float*)d_in[4];
  const float* W2   = (const float*)d_in[5];
  const float* b2   = (const float*)d_in[6];
  const float* Wfc  = (const float*)d_in[7];
  const float* bfc  = (const float*)d_in[8];
  float* out = (float*)d_out;

  const int N = in_sizes[0] / F_IN;     // 100000
  const int E = in_sizes[1] / 2;        // 400000
  const int G = out_size / OUT_F;       // 2048
  const int* src = eidx;
  const int* dst = eidx + E;

  // workspace carve-up (bytes, 256-aligned)
  char* ws = (char*)d_ws;
  size_t off = 0;
  auto take = [&](size_t bytes) { char* p = ws + off; off = (off + bytes + 255) & ~(size_t)255; return p; };
  float* dinv   = (float*)take((size_t)N * 4);
  float* h      = (float*)take((size_t)N * HID * 4);
  float* agg    = (float*)take((size_t)N * HID * 4);
  float* pooled = (float*)take((size_t)G * HID * 4);
  float* cnt    = (float*)take((size_t)G * 4);
  (void)ws_size;

  const int TB = 256;
  const long long nodeFeat = (long long)N * 64;   // float4-chunk threads per node pass
  const long long edgeFeat = (long long)E * 64;

  // degrees -> dinv
  zero_kernel<<<ceil_div(N, TB), TB, 0, stream>>>(dinv, N);
  deg_edges_kernel<<<ceil_div(E, TB), TB, 0, stream>>>(dst, dinv, E);
  dinv_kernel<<<ceil_div(N, TB), TB, 0, stream>>>(dinv, N);

  // layer 1: h = x @ W1 ; agg = scatter(h) ; relu(agg + b1)
  wmma_gemm_kernel<<<N / 32, (HID / 64) * 32, 0, stream>>>(x, W1, h, nullptr, N, HID, F_IN);
  self_init_kernel<<<ceil_div(nodeFeat, TB), TB, 0, stream>>>(h, dinv, agg, N);
  edge_scatter_kernel<<<ceil_div(edgeFeat, TB), TB, 0, stream>>>(h, dinv, src, dst, agg, E);
  bias_relu_kernel<<<ceil_div((long long)N * HID, TB), TB, 0, stream>>>(agg, b1, N * HID);

  // layer 2: h = agg @ W2 ; agg = scatter(h) ; relu(agg + b2)
  wmma_gemm_kernel<<<N / 32, (HID / 64) * 32, 0, stream>>>(agg, W2, h, nullptr, N, HID, HID);
  self_init_kernel<<<ceil_div(nodeFeat, TB), TB, 0, stream>>>(h, dinv, agg, N);
  edge_scatter_kernel<<<ceil_div(edgeFeat, TB), TB, 0, stream>>>(h, dinv, src, dst, agg, E);
  bias_relu_kernel<<<ceil_div((long long)N * HID, TB), TB, 0, stream>>>(agg, b2, N * HID);

  // global mean pool
  zero_kernel<<<ceil_div((long long)G * HID, TB), TB, 0, stream>>>(pooled, G * HID);
  zero_kernel<<<ceil_div(G, TB), TB, 0, stream>>>(cnt, G);
  pool_scatter_kernel<<<ceil_div(nodeFeat, TB), TB, 0, stream>>>(agg, batch, pooled, cnt, N);
  pool_div_kernel<<<ceil_div((long long)G * HID, TB), TB, 0, stream>>>(pooled, cnt, G * HID);

  // FC: out = pooled @ Wfc + bfc   (N=128 -> 2 waves per block)
  wmma_gemm_kernel<<<G / 32, (OUT_F / 64) * 32, 0, stream>>>(pooled, Wfc, out, bfc, G, OUT_F, HID);
}